// Kernel_7988639170861
// MI455X (gfx1250) — compile-verified
//
#include <hip/hip_runtime.h>
#include <math.h>

typedef __attribute__((ext_vector_type(2))) float v2f;
typedef __attribute__((ext_vector_type(8))) float v8f;

// ---- problem constants ----
#define BATCH   16
#define CF      32          // feature channels
#define FHW     64          // feature H=W
#define OC      16          // conv output channels
#define KTOT    8192        // 32*16*16 reduction
#define NSPLIT  8           // K-split across blocks for stage1a
#define KCHUNK  (KTOT / NSPLIT)   // 1024
#define CONVN   144         // 16 oc * 9 spatial
#define IMGC    3
#define IMGHW   512

// workspace layout (floats)
#define WS_PARTIAL 0                         // NSPLIT * 2304
#define WS_KV      (NSPLIT * BATCH * CONVN)  // 18432 : 16*25 kernel values

// ============================================================
// Stage 1a: conv-as-GEMM via V_WMMA_F32_16X16X4_F32.
// grid = NSPLIT blocks, block = 288 threads (9 waves, 1 per (oy,ox)).
// Each wave computes C[b=0..15][oc=0..15] for its position over a K chunk.
// ============================================================
__global__ __launch_bounds__(288) void stage1a_gemm(
    const float* __restrict__ feature,   // (16,32,64,64)
    const float* __restrict__ conv_w,    // (16,32,16,16)
    float* __restrict__ ws_partial)      // (NSPLIT, 16, 144)
{
  const int tid  = threadIdx.x;
  const int wave = tid >> 5;        // 0..8  -> spatial position
  const int lane = tid & 31;
  const int m    = lane & 15;       // A row (batch) / B col (oc)
  const int kh   = lane >> 4;       // which K-half this lane supplies
  const int oy   = wave / 3;
  const int ox   = wave - 3 * oy;

  const int kbeg = blockIdx.x * KCHUNK;
  const int kend = kbeg + KCHUNK;

  // base offset of feature[m][0][oy-1][ox-1] (may be "negative"; only used
  // when masked valid, loads made unconditional via clamped index)
  const int mbase = m * (CF * FHW * FHW) + (oy - 1) * FHW + (ox - 1);
  const float* __restrict__ wrow = conv_w + m * KTOT;   // B operand: oc = m

  v8f acc = {};
  for (int k0 = kbeg; k0 < kend; k0 += 4) {
    const int k  = k0 + 2 * kh;        // even, kx in {0,2,...,14}
    const int ic = k >> 8;
    const int ky = (k >> 4) & 15;
    const int kx = k & 15;
    const int iy = oy - 1 + ky;
    const int ix = ox - 1 + kx;
    const int foff = (ic << 12) + (ky << 6) + kx;  // ic*4096 + ky*64 + kx

    const bool vy = ((unsigned)iy < 16u);
    const bool v0 = vy && ((unsigned)ix < 16u);
    const bool v1 = vy && ((unsigned)(ix + 1) < 16u);

    // unconditional loads with clamped-safe indices, then select
    const int i0 = v0 ? (mbase + foff)     : 0;
    const int i1 = v1 ? (mbase + foff + 1) : 0;
    const float f0 = feature[i0];
    const float f1 = feature[i1];

    v2f a, b;
    a.x = v0 ? f0 : 0.0f;
    a.y = v1 ? f1 : 0.0f;
    b.x = wrow[k];
    b.y = wrow[k + 1];

    acc = __builtin_amdgcn_wmma_f32_16x16x4_f32(
        /*neg_a=*/false, a, /*neg_b=*/false, b,
        /*c_mod=*/(short)0, acc, /*reuse_a=*/false, /*reuse_b=*/false);
  }

  // C/D layout: VGPR r, lanes 0-15 -> M=r, N=lane ; lanes 16-31 -> M=r+8, N=lane-16
  float* __restrict__ dst = ws_partial + blockIdx.x * (BATCH * CONVN);
#pragma unroll
  for (int r = 0; r < 8; ++r) {
    const int bb = r + 8 * kh;                 // batch index
    dst[bb * CONVN + m * 9 + wave] = acc[r];   // n = oc*9 + (oy*3+ox)
  }
}

// ============================================================
// Stage 1b: reduce K-splits, + bias, exact GELU, linear 144->25.
// single block of 288 threads.
// ============================================================
__global__ __launch_bounds__(288) void stage1b_head(
    const float* __restrict__ ws_partial,
    const float* __restrict__ conv_b,    // (16,)
    const float* __restrict__ lin_w,     // (25,144)
    const float* __restrict__ lin_b,     // (25,)
    float* __restrict__ kv_out)          // (16,25)
{
  __shared__ float sfeat[BATCH * CONVN];       // 2304 floats
  const int tid = threadIdx.x;

  for (int i = tid; i < BATCH * CONVN; i += 288) {
    float s = 0.0f;
#pragma unroll
    for (int sp = 0; sp < NSPLIT; ++sp)
      s += ws_partial[sp * (BATCH * CONVN) + i];
    const int oc = (i % CONVN) / 9;
    const float x = s + conv_b[oc];
    // exact GELU: 0.5*x*(1+erf(x/sqrt(2)))
    sfeat[i] = 0.5f * x * (1.0f + erff(x * 0.70710678118654752f));
  }
  __syncthreads();

  for (int idx = tid; idx < BATCH * 25; idx += 288) {
    const int bb = idx / 25;
    const int j  = idx - 25 * bb;
    float s = lin_b[j];
    const float* __restrict__ fr = sfeat + bb * CONVN;
    const float* __restrict__ wr = lin_w + j * CONVN;
#pragma unroll 8
    for (int n = 0; n < CONVN; ++n)
      s = fmaf(fr[n], wr[n], s);
    kv_out[bb * 25 + j] = s;
  }
}

// ============================================================
// Stage 2: per-image 5x5 depthwise conv with reflect padding.
// 32x32 output tile / block, 36x36 LDS input tile.
// Each thread: full 8x5 register window, 4 outputs in a column
// computed with 4 INDEPENDENT accumulators so every tap issues
// 4 parallel FMAs (VOPD-pairable), no serial 25-deep chains.
// grid = (16, 16, 48), block = (32, 8)
// ============================================================
#define TILE 32
#define LW   36

__device__ __forceinline__ int reflect512(int t) {
  t = (t < 0) ? -t : t;
  return (t > 511) ? (1022 - t) : t;
}

__global__ __launch_bounds__(256) void stage2_dwconv(
    const float* __restrict__ img,   // (16,3,512,512)
    const float* __restrict__ kv,    // (16,25)
    float* __restrict__ out)         // (16,3,512,512)
{
  __shared__ float tile[LW * LW];

  const int plane = blockIdx.z;          // b*3 + c
  const int b = plane / IMGC;
  const float* __restrict__ src = img + (size_t)plane * IMGHW * IMGHW;
  float* __restrict__ dst       = out + (size_t)plane * IMGHW * IMGHW;

  // per-image 5x5 kernel; uniform address -> scalar loads -> SGPR operands
  float w[25];
  const float* __restrict__ kb = kv + b * 25;
#pragma unroll
  for (int i = 0; i < 25; ++i) w[i] = kb[i];

  const int tx0 = blockIdx.x * TILE - 2;
  const int ty0 = blockIdx.y * TILE - 2;
  const int tid = threadIdx.y * 32 + threadIdx.x;

  for (int i = tid; i < LW * LW; i += 256) {
    const int ly = i / LW;
    const int lx = i - LW * ly;
    const int gy = reflect512(ty0 + ly);
    const int gx = reflect512(tx0 + lx);
    tile[i] = src[gy * IMGHW + gx];
  }
  __syncthreads();

  const int tx = threadIdx.x;
  const int y0 = threadIdx.y * 4;   // this thread's 4 output rows: y0..y0+3

  // full window: rows y0..y0+7, cols tx..tx+4
  float win[8][5];
#pragma unroll
  for (int r = 0; r < 8; ++r)
#pragma unroll
    for (int x = 0; x < 5; ++x)
      win[r][x] = tile[(y0 + r) * LW + tx + x];

  float acc0 = 0.0f, acc1 = 0.0f, acc2 = 0.0f, acc3 = 0.0f;
#pragma unroll
  for (int j = 0; j < 5; ++j) {
#pragma unroll
    for (int x = 0; x < 5; ++x) {
      const float wv = w[j * 5 + x];
      acc0 = fmaf(wv, win[j + 0][x], acc0);
      acc1 = fmaf(wv, win[j + 1][x], acc1);
      acc2 = fmaf(wv, win[j + 2][x], acc2);
      acc3 = fmaf(wv, win[j + 3][x], acc3);
    }
  }

  const int gx  = blockIdx.x * TILE + tx;
  const int gyb = blockIdx.y * TILE + y0;
  dst[(size_t)(gyb + 0) * IMGHW + gx] = acc0;
  dst[(size_t)(gyb + 1) * IMGHW + gx] = acc1;
  dst[(size_t)(gyb + 2) * IMGHW + gx] = acc2;
  dst[(size_t)(gyb + 3) * IMGHW + gx] = acc3;
}

// ============================================================
extern "C" void kernel_launch(void* const* d_in, const int* in_sizes, int n_in,
                              void* d_out, int out_size, void* d_ws, size_t ws_size,
                              hipStream_t stream) {
  const float* feature = (const float*)d_in[0];  // (16,32,64,64)
  const float* img     = (const float*)d_in[1];  // (16,3,512,512)
  const float* conv_w  = (const float*)d_in[2];  // (16,32,16,16)
  const float* conv_b  = (const float*)d_in[3];  // (16,)
  const float* lin_w   = (const float*)d_in[4];  // (25,144)
  const float* lin_b   = (const float*)d_in[5];  // (25,)
  float* out = (float*)d_out;
  float* ws  = (float*)d_ws;

  float* ws_partial = ws + WS_PARTIAL;
  float* kv         = ws + WS_KV;

  stage1a_gemm<<<dim3(NSPLIT), dim3(288), 0, stream>>>(feature, conv_w, ws_partial);
  stage1b_head<<<dim3(1), dim3(288), 0, stream>>>(ws_partial, conv_b, lin_w, lin_b, kv);
  stage2_dwconv<<<dim3(IMGHW / TILE, IMGHW / TILE, BATCH * IMGC),
                  dim3(32, 8), 0, stream>>>(img, kv, out);
}